// ArchMaster_51092930953535
// MI455X (gfx1250) — compile-verified
//
#include <hip/hip_runtime.h>
#include <math.h>

#define N_NODES 5
#define N_OPS   8
#define HID     100
#define VOCAB   (N_NODES + 1 + N_OPS)   // 14
#define TEMP            5.0f
#define TANH_C          2.5f
#define OP_TANH_REDUCE  2.5f
#define EW      112                     // padded Eattn row stride (7*16)
#define NT      512                     // 16 wave32s on one WGP

typedef __attribute__((ext_vector_type(2))) float v2f;
typedef __attribute__((ext_vector_type(8))) float v8f;

__device__ __forceinline__ float sigmoidf_(float x) { return 1.0f / (1.0f + expf(-x)); }

__global__ __launch_bounds__(NT) void nas_ctrl_kernel(
    const float* __restrict__ emb,   const float* __restrict__ Wea,
    const float* __restrict__ Wha,   const float* __restrict__ vat,
    const float* __restrict__ Wsoft, const float* __restrict__ bsoft,
    const float* __restrict__ Wih,   const float* __restrict__ Whh,
    const float* __restrict__ bih,   const float* __restrict__ bhh,
    const float* __restrict__ u,     float* __restrict__ out)
{
  const int tid  = threadIdx.x;
  const int lane = tid & 31;
  const int wv   = tid >> 5;

  __shared__ __align__(16) float sE[6 * EW];      // Eattn = emb[:6] @ Wea^T
  __shared__ __align__(16) float sEmb[16 * HID];  // zero-padded A staging (rows >= VOCAB -> 0)
  __shared__ __align__(16) float sWpad[EW * HID]; // zero-padded B staging (rows >= HID -> 0)
  __shared__ __align__(16) float sx[HID], sh[HID], sc[HID];
  __shared__ __align__(16) float sg[4 * HID];     // LSTM gates; reused as h@Wha^T scratch
  __shared__ float satt[N_OPS];                   // logits (<= 8)
  __shared__ int   s_inp;

  for (int t = tid; t < HID; t += NT) { sh[t] = 0.f; sc[t] = 0.f; sx[t] = 0.f; }
  if (tid == 0) s_inp = 0;

  // ---- stage zero-padded WMMA operand tiles into LDS (coalesced, guard-free WMMA loop) ----
  for (int idx = tid; idx < 16 * HID; idx += NT) {
    const int r = idx / HID, c = idx % HID;
    sEmb[idx] = (r < VOCAB) ? emb[r * HID + c] : 0.f;
  }
  for (int idx = tid; idx < EW * HID; idx += NT) {
    const int r = idx / HID, c = idx % HID;
    sWpad[idx] = (r < HID) ? Wea[r * HID + c] : 0.f;
  }
  __syncthreads();

  // ---- Phase A: Eattn via V_WMMA_F32_16X16X4_F32; 7 N-tiles on 7 waves in parallel ----
#if __has_builtin(__builtin_amdgcn_wmma_f32_16x16x4_f32)
  if (wv < 7) {
    const int nt = wv;
    const int m  = lane & 15;          // A row (M)
    const int kb = (lane >> 4) * 2;    // K sub-offset per ISA 16x4-f32 layout
    const int n  = nt * 16 + (lane & 15);   // B column (N), padded rows are zero
    v8f acc = {0.f, 0.f, 0.f, 0.f, 0.f, 0.f, 0.f, 0.f};
    #pragma unroll
    for (int kt = 0; kt < 25; ++kt) {
      const int k = kt * 4 + kb;       // even -> 8B-aligned ds_load_b64
      const v2f a = *(const v2f*)&sEmb[m * HID + k];
      const v2f b = *(const v2f*)&sWpad[n * HID + k];   // B[k][n] = Wea[n][k]
      acc = __builtin_amdgcn_wmma_f32_16x16x4_f32(
          false, a, false, b, (short)0, acc, false, false);
    }
    if (lane < 16) {                   // D rows 0..5 live in VGPRs 0..5, lanes 0-15
      #pragma unroll
      for (int r = 0; r < 6; ++r) sE[r * EW + nt * 16 + lane] = acc[r];
    }
  }
#else
  for (int idx = tid; idx < 6 * HID; idx += NT) {
    const int r = idx / HID, cc = idx % HID;
    float a = 0.f;
    for (int k = 0; k < HID; ++k) a = fmaf(sEmb[r * HID + k], sWpad[cc * HID + k], a);
    sE[r * EW + cc] = a;
  }
#endif
  __syncthreads();

  float log_p = 0.f, ent = 0.f;   // thread 0 authoritative
  int step = 0;

  for (int node = 0; node < N_NODES; ++node) {
    for (int ph = 0; ph < 4; ++ph) {
      const bool is_node = (ph < 2);

      // ---- input embedding ----
      if (!(node == 0 && ph == 0)) {
        const int ii = s_inp;          // written by thread 0 before trailing sync
        for (int t = tid; t < HID; t += NT) sx[t] = emb[ii * HID + t];
      }
      __syncthreads();

      // ---- LSTM cell: g = W_ih x + b_ih + W_hh h + b_hh (float4 / b128 loads) ----
      for (int j = tid; j < 4 * HID; j += NT) {
        float acc = bih[j] + bhh[j];
        const float4* wi = (const float4*)(Wih + j * HID);  // rows are 400B, 16B-aligned
        const float4* wh = (const float4*)(Whh + j * HID);
        const float4* x4 = (const float4*)sx;
        const float4* h4 = (const float4*)sh;
        #pragma unroll 5
        for (int k = 0; k < HID / 4; ++k) {
          const float4 a = wi[k], b = wh[k], xv = x4[k], hv = h4[k];
          acc = fmaf(a.x, xv.x, fmaf(a.y, xv.y, fmaf(a.z, xv.z, fmaf(a.w, xv.w, acc))));
          acc = fmaf(b.x, hv.x, fmaf(b.y, hv.y, fmaf(b.z, hv.z, fmaf(b.w, hv.w, acc))));
        }
        sg[j] = acc;
      }
      __syncthreads();
      for (int t = tid; t < HID; t += NT) {
        const float ig = sigmoidf_(sg[t]);
        const float fg = sigmoidf_(sg[HID + t]);
        const float gg = tanhf(sg[2 * HID + t]);
        const float og = sigmoidf_(sg[3 * HID + t]);
        const float c2 = fg * sc[t] + ig * gg;
        sc[t] = c2;
        sh[t] = og * tanhf(c2);
      }
      __syncthreads();

      int nAct;
      if (is_node) {
        // hw = Wha @ h  -> sg[0..99] (gates consumed, buffer free)
        for (int t = tid; t < HID; t += NT) {
          float acc = 0.f;
          const float4* wr = (const float4*)(Wha + t * HID);
          const float4* h4 = (const float4*)sh;
          #pragma unroll 5
          for (int k = 0; k < HID / 4; ++k) {
            const float4 a = wr[k], hv = h4[k];
            acc = fmaf(a.x, hv.x, fmaf(a.y, hv.y, fmaf(a.z, hv.z, fmaf(a.w, hv.w, acc))));
          }
          sg[t] = acc;
        }
        __syncthreads();
        nAct = node + 2;
        if (wv < nAct) {               // one wave per candidate row
          float s = 0.f;
          for (int t = lane; t < HID; t += 32)
            s += vat[t] * tanhf(sE[wv * EW + t] + sg[t]);
          for (int off = 16; off > 0; off >>= 1) s += __shfl_down(s, off, 32);
          if (lane == 0) satt[wv] = s;
        }
      } else {
        nAct = N_OPS;
        if (wv < N_OPS) {              // one wave per op logit
          float s = 0.f;
          for (int t = lane; t < HID; t += 32)
            s += Wsoft[wv * HID + t] * sh[t];
          for (int off = 16; off > 0; off >>= 1) s += __shfl_down(s, off, 32);
          if (lane == 0) satt[wv] = s + bsoft[wv];
        }
      }
      __syncthreads();

      // ---- softmax + inverse-CDF sample (tiny, thread 0) ----
      if (tid == 0) {
        const float scale = is_node ? TANH_C : (TANH_C / OP_TANH_REDUCE);
        float lg[N_OPS];
        float mx = -1e30f;
        for (int m2 = 0; m2 < nAct; ++m2) {
          lg[m2] = scale * tanhf(satt[m2] / TEMP);
          mx = fmaxf(mx, lg[m2]);
        }
        float se = 0.f;
        for (int m2 = 0; m2 < nAct; ++m2) se += expf(lg[m2] - mx);
        const float lse = mx + logf(se);
        const float uu  = u[step];
        float csum = 0.f; int act = 0;
        for (int m2 = 0; m2 < nAct; ++m2) {
          const float p  = expf(lg[m2] - lse);
          const float lp = lg[m2] - lse;
          csum += p;
          if (csum < uu) ++act;
          ent -= p * lp;
        }
        if (act > nAct - 1) act = nAct - 1;
        log_p += lg[act] - lse;
        if (is_node) {
          out[node * 2 + ph] = (float)act;
          s_inp = act;
        } else {
          out[10 + node * 2 + (ph - 2)] = (float)act;
          s_inp = act + N_NODES + 1;
        }
      }
      ++step;
      __syncthreads();
    }
  }
  if (tid == 0) { out[20] = log_p; out[21] = ent; }
}

extern "C" void kernel_launch(void* const* d_in, const int* in_sizes, int n_in,
                              void* d_out, int out_size, void* d_ws, size_t ws_size,
                              hipStream_t stream) {
  (void)in_sizes; (void)n_in; (void)out_size; (void)d_ws; (void)ws_size;
  const float* emb   = (const float*)d_in[0];
  const float* Wea   = (const float*)d_in[1];
  const float* Wha   = (const float*)d_in[2];
  const float* vat   = (const float*)d_in[3];
  const float* Wsoft = (const float*)d_in[4];
  const float* bsoft = (const float*)d_in[5];
  const float* Wih   = (const float*)d_in[6];
  const float* Whh   = (const float*)d_in[7];
  const float* bih   = (const float*)d_in[8];
  const float* bhh   = (const float*)d_in[9];
  const float* u     = (const float*)d_in[10];
  nas_ctrl_kernel<<<dim3(1), dim3(NT), 0, stream>>>(
      emb, Wea, Wha, vat, Wsoft, bsoft, Wih, Whh, bih, bhh, u, (float*)d_out);
}